// My_LightGCN_4114578669910
// MI455X (gfx1250) — compile-verified
//
#include <hip/hip_runtime.h>

// ---------------- problem constants ----------------
#define NODES 200000
#define EDGES 1250000
#define DIM   64
#define NV    (NODES * DIM)        // 12,800,000 floats per node-feature buffer
#define PAIRS (4096 * 101)         // 413,696
#define TILES (PAIRS / 16)         // 25,856 (exact)

typedef __attribute__((ext_vector_type(2))) float v2f;
typedef __attribute__((ext_vector_type(4))) float v4f;
typedef __attribute__((ext_vector_type(8))) float v8f;

// ---------------- utility kernels ----------------
__global__ void zero_f4(float* __restrict__ p, int n4) {
  int i = blockIdx.x * blockDim.x + threadIdx.x;
  if (i < n4) ((v4f*)p)[i] = (v4f){0.f, 0.f, 0.f, 0.f};
}

__global__ void copy_f4(const float* __restrict__ src, float* __restrict__ dst, int n4) {
  int i = blockIdx.x * blockDim.x + threadIdx.x;
  if (i < n4) ((v4f*)dst)[i] = ((const v4f*)src)[i];
}

__global__ void accum_f4(float* __restrict__ acc, const float* __restrict__ x, int n4) {
  int i = blockIdx.x * blockDim.x + threadIdx.x;
  if (i < n4) {
    v4f a = ((v4f*)acc)[i];
    v4f b = ((const v4f*)x)[i];
    ((v4f*)acc)[i] = a + b;
  }
}

// ---------------- graph normalization ----------------
__global__ void degree_k(const int* __restrict__ col, float* __restrict__ deg) {
  int e = blockIdx.x * blockDim.x + threadIdx.x;
  if (e < EDGES) unsafeAtomicAdd(&deg[col[e]], 1.0f);
}

__global__ void rsqrt_k(float* __restrict__ deg) {
  int i = blockIdx.x * blockDim.x + threadIdx.x;
  if (i < NODES) {
    float d = deg[i];
    deg[i] = (d > 0.f) ? rsqrtf(d) : 0.f;   // deg^{-1/2}, isolated -> 0
  }
}

__global__ void norm_k(const int* __restrict__ row, const int* __restrict__ col,
                       const float* __restrict__ dis, float* __restrict__ nrm) {
  int e = blockIdx.x * blockDim.x + threadIdx.x;
  if (e < EDGES) nrm[e] = dis[row[e]] * dis[col[e]];
}

// ---------------- SpMM scatter: one wave32 per edge ----------------
// lane l handles dims l and l+32 -> two fully-coalesced 128B atomic bursts.
// All feature buffers (~154MB) fit in the 192MB L2, atomics execute there.
__global__ void scatter_k(const int* __restrict__ row, const int* __restrict__ col,
                          const float* __restrict__ nrm,
                          const float* __restrict__ xin, float* __restrict__ xout) {
  int wave = (blockIdx.x * blockDim.x + threadIdx.x) >> 5;
  int lane = threadIdx.x & 31;
  if (wave < EDGES) {
    int s = row[wave];
    int d = col[wave];
    float w = nrm[wave];
    const float* xs = xin + (size_t)s * DIM;
    float*       xd = xout + (size_t)d * DIM;
    unsafeAtomicAdd(&xd[lane],      w * xs[lane]);
    unsafeAtomicAdd(&xd[lane + 32], w * xs[lane + 32]);
  }
}

// ---------------- scoring: V_WMMA_F32_16X16X4_F32, diagonal = 16 logits ----
// A-matrix (16x4 f32): lane l holds A[l&15][2*(l>>4)+{0,1}] -> float2 load.
// B-matrix (4x16 f32): same per-lane addressing on the item row.
// Chain 16 WMMAs over K=64 in full f32 (no precision loss vs reference).
// alpha = 1/4 folded in as 1/16 on the final dot product.
__global__ void score_k(const float* __restrict__ acc, const int* __restrict__ batch,
                        float* __restrict__ out) {
  int wid  = (blockIdx.x * blockDim.x + threadIdx.x) >> 5;   // tile of 16 pairs
  int lane = threadIdx.x & 31;

  int m     = lane & 15;            // pair within tile (both half-waves)
  int kbase = (lane >> 4) * 2;      // K sub-offset for this half-wave
  int p     = wid * 16 + m;

  int u  = batch[p * 3 + 0];
  int it = batch[p * 3 + 1];
  const float* ur = acc + (size_t)u  * DIM + kbase;
  const float* ir = acc + (size_t)it * DIM + kbase;

  v8f c = {};
#pragma unroll
  for (int s = 0; s < 16; ++s) {
    v2f a = *(const v2f*)(ur + 4 * s);
    v2f b = *(const v2f*)(ir + 4 * s);
    c = __builtin_amdgcn_wmma_f32_16x16x4_f32(
        /*neg_a=*/false, a, /*neg_b=*/false, b,
        /*c_mod=*/(short)0, c, /*reuse_a=*/false, /*reuse_b=*/false);
  }

  // D[i][i]: i<8 -> lane i, vgpr i ; i>=8 -> lane i+16, vgpr i-8
  bool active = (lane < 8) || (lane >= 24);
  int  d      = (lane < 8) ? lane : lane - 24;
  int  iout   = (lane < 8) ? lane : lane - 16;

  float val = c[0];
  if (d == 1) val = c[1];
  if (d == 2) val = c[2];
  if (d == 3) val = c[3];
  if (d == 4) val = c[4];
  if (d == 5) val = c[5];
  if (d == 6) val = c[6];
  if (d == 7) val = c[7];

  if (active) out[wid * 16 + iout] = val * 0.0625f;   // alpha^2 = 1/16
}

// ---------------- host orchestration ----------------
extern "C" void kernel_launch(void* const* d_in, const int* in_sizes, int n_in,
                              void* d_out, int out_size, void* d_ws, size_t ws_size,
                              hipStream_t stream) {
  const float* emb = (const float*)d_in[0];
  const int*   ei  = (const int*)d_in[1];
  const int*   bat = (const int*)d_in[2];
  const int* row = ei;             // edge_index[0]
  const int* col = ei + EDGES;     // edge_index[1]
  float* out = (float*)d_out;

  float* ws  = (float*)d_ws;
  float* acc = ws;                       // NV floats
  float* xA  = ws + (size_t)NV;          // NV floats
  float* xB  = ws + (size_t)2 * NV;      // NV floats
  float* deg = ws + (size_t)3 * NV;      // NODES floats (reused as deg^{-1/2})
  float* nrm = deg + NODES;              // EDGES floats

  const int T = 256;
  const int nv4   = NV / 4;                       // 3,200,000
  const int gNV4  = (nv4 + T - 1) / T;            // node-feature f4 grid
  const int gDeg4 = (NODES / 4 + T - 1) / T;
  const int gNode = (NODES + T - 1) / T;
  const int gEdge = (EDGES + T - 1) / T;
  const int gScat = (EDGES * 32 + T - 1) / T;     // one wave per edge
  const int gScore = TILES / (T / 32);            // 25,856 / 8 = 3,232 exact

  // init: acc = emb, deg = 0, xA = 0
  copy_f4<<<gNV4, T, 0, stream>>>(emb, acc, nv4);
  zero_f4<<<gDeg4, T, 0, stream>>>(deg, NODES / 4);
  zero_f4<<<gNV4, T, 0, stream>>>(xA, nv4);

  // normalization coefficients
  degree_k<<<gEdge, T, 0, stream>>>(col, deg);
  rsqrt_k<<<gNode, T, 0, stream>>>(deg);
  norm_k<<<gEdge, T, 0, stream>>>(row, col, deg, nrm);

  // layer 1: emb -> xA
  scatter_k<<<gScat, T, 0, stream>>>(row, col, nrm, emb, xA);
  accum_f4<<<gNV4, T, 0, stream>>>(acc, xA, nv4);
  zero_f4<<<gNV4, T, 0, stream>>>(xB, nv4);

  // layer 2: xA -> xB
  scatter_k<<<gScat, T, 0, stream>>>(row, col, nrm, xA, xB);
  accum_f4<<<gNV4, T, 0, stream>>>(acc, xB, nv4);
  zero_f4<<<gNV4, T, 0, stream>>>(xA, nv4);

  // layer 3: xB -> xA
  scatter_k<<<gScat, T, 0, stream>>>(row, col, nrm, xB, xA);
  accum_f4<<<gNV4, T, 0, stream>>>(acc, xA, nv4);

  // scoring (WMMA f32 16x16x4)
  score_k<<<gScore, T, 0, stream>>>(acc, bat, out);
}